// WiredLayer_45423574122556
// MI455X (gfx1250) — compile-verified
//
#include <hip/hip_runtime.h>

typedef __attribute__((ext_vector_type(2))) float v2f;
typedef __attribute__((ext_vector_type(8))) float v8f;

#define N_NODES 10
#define DIM 64
#define TM 64                       // batch rows per workgroup
#define PITCH 68                    // pitched LDS row (floats) -> bank-conflict-free
#define NODE_STRIDE (TM * PITCH)    // floats per staged node tile
#define LDS_FLOATS (9 * NODE_STRIDE)

// ---------------------------------------------------------------------------
// One-time (per call) weight transpose: Wt[e][n][k] = W[e][k][n], e = f*10+t.
// Makes the WMMA B-fragment (two k-consecutive elements per lane) one b64 load.
// ---------------------------------------------------------------------------
__global__ __launch_bounds__(256) void transpose_w(const float* __restrict__ W,
                                                   float* __restrict__ Wt) {
  __shared__ float tile[DIM][DIM + 1];
  const int e = blockIdx.x;
  const float* src = W + (size_t)e * DIM * DIM;
  float* dst = Wt + (size_t)e * DIM * DIM;
  for (int i = threadIdx.x; i < DIM * DIM; i += 256)
    tile[i >> 6][i & 63] = src[i];
  __syncthreads();
  for (int i = threadIdx.x; i < DIM * DIM; i += 256) {
    const int n = i >> 6, k = i & 63;
    dst[i] = tile[k][n];
  }
}

// ---------------------------------------------------------------------------
// Fused DAG kernel: 4 wave32 waves / block, 64 batch rows, all intermediate
// activations live in LDS. Each wave owns a 2x2 block of 16x16 C tiles so
// every A/B fragment feeds two v_wmma_f32_16x16x4_f32.
// ---------------------------------------------------------------------------
template <bool XPOSED>
__global__ __launch_bounds__(128) void wired_dag_wmma(
    const float* __restrict__ x, const float* __restrict__ Wm,
    const float* __restrict__ bias, float* __restrict__ out)
{
  extern __shared__ __align__(16) float smem[];
  const int tid  = threadIdx.x;
  const int lane = tid & 31;
  const int wave = tid >> 5;                // 0..3
  const long rowBase = (long)blockIdx.x * TM;

  // Stage x (node 0) into pitched LDS tile with float4 loads.
  {
    const float4* src = (const float4*)(x + rowBase * DIM);
    for (int i = tid; i < (TM * DIM) / 4; i += 128) {
      const int r  = i >> 4;          // row 0..63
      const int c4 = (i & 15) << 2;   // col 0,4,...,60
      *(float4*)&smem[r * PITCH + c4] = src[i];
    }
  }
  __syncthreads();

  // 2x2 C-tile block per wave: row tiles {mta, mta+1}, col tiles {ntp*2, ntp*2+1}.
  const int mta  = (wave >> 1) * 2;         // 0 or 2
  const int ntp  = wave & 1;                // 0 or 1
  const int l16  = lane & 15;
  const int half = lane >> 4;               // 0/1 (lane group)
  const int arowA = mta * 16 + l16;         // A rows for tile-row a
  const int col0  = ntp * 32 + l16;         // cols for tile-col 0
  const int col1  = col0 + 16;              // cols for tile-col 1

  for (int t = 1; t < N_NODES; ++t) {
    // C init = summed bias (column-only -> one scalar per lane per tile-col).
    float b0 = 0.f, b1 = 0.f;
    for (int f = 0; f < t; ++f) {
      const float* bp = bias + (size_t)(f * N_NODES + t) * DIM;
      b0 += bp[col0];
      b1 += bp[col1];
    }
    v8f c00, c01, c10, c11;
    #pragma unroll
    for (int i = 0; i < 8; ++i) {
      c00[i] = b0; c10[i] = b0;
      c01[i] = b1; c11[i] = b1;
    }

    for (int f = 0; f < t; ++f) {
      const float* apA = &smem[f * NODE_STRIDE + arowA * PITCH];
      const float* apB = apA + 16 * PITCH;
      const float* wp  = Wm + (size_t)(f * N_NODES + t) * DIM * DIM;
      #pragma unroll
      for (int k0 = 0; k0 < DIM; k0 += 4) {
        const int kk = k0 + half * 2;       // ISA A/B layout: K split by lane half
        v2f a0 = *(const v2f*)(apA + kk);
        v2f a1 = *(const v2f*)(apB + kk);
        v2f p, q;
        if (XPOSED) {                        // Wt[n][k]: contiguous b64 per lane
          p = *(const v2f*)(wp + col0 * DIM + kk);
          q = *(const v2f*)(wp + col1 * DIM + kk);
        } else {                             // W[k][n]: two strided b32 per lane
          p.x = wp[kk * DIM + col0]; p.y = wp[(kk + 1) * DIM + col0];
          q.x = wp[kk * DIM + col1]; q.y = wp[(kk + 1) * DIM + col1];
        }
        c00 = __builtin_amdgcn_wmma_f32_16x16x4_f32(false, a0, false, p, (short)0, c00, false, false);
        c01 = __builtin_amdgcn_wmma_f32_16x16x4_f32(false, a0, false, q, (short)0, c01, false, false);
        c10 = __builtin_amdgcn_wmma_f32_16x16x4_f32(false, a1, false, p, (short)0, c10, false, false);
        c11 = __builtin_amdgcn_wmma_f32_16x16x4_f32(false, a1, false, q, (short)0, c11, false, false);
      }
    }

    if (t < N_NODES - 1) {
      // Store act_t into its LDS tile (C layout: VGPR i -> row half*8+i).
      float* dp = &smem[t * NODE_STRIDE];
      #pragma unroll
      for (int i = 0; i < 8; ++i) {
        const int ra = mta * 16 + half * 8 + i;
        const int rb = ra + 16;
        dp[ra * PITCH + col0] = c00[i];
        dp[ra * PITCH + col1] = c01[i];
        dp[rb * PITCH + col0] = c10[i];
        dp[rb * PITCH + col1] = c11[i];
      }
      __syncthreads();
    } else {
      // Output node: write straight to global.
      float* op = out + rowBase * DIM;
      #pragma unroll
      for (int i = 0; i < 8; ++i) {
        const int ra = mta * 16 + half * 8 + i;
        const int rb = ra + 16;
        op[(size_t)ra * DIM + col0] = c00[i];
        op[(size_t)ra * DIM + col1] = c01[i];
        op[(size_t)rb * DIM + col0] = c10[i];
        op[(size_t)rb * DIM + col1] = c11[i];
      }
    }
  }
}

extern "C" void kernel_launch(void* const* d_in, const int* in_sizes, int n_in,
                              void* d_out, int out_size, void* d_ws, size_t ws_size,
                              hipStream_t stream) {
  const float* x  = (const float*)d_in[0];
  const float* W  = (const float*)d_in[1];
  const float* b  = (const float*)d_in[2];
  float* out      = (float*)d_out;

  const int Bn      = in_sizes[0] / DIM;   // 65536
  const int nBlocks = Bn / TM;             // 1024
  const size_t ldsBytes = (size_t)LDS_FLOATS * sizeof(float); // 156672 B
  const size_t wtBytes  = (size_t)N_NODES * N_NODES * DIM * DIM * sizeof(float);

  if (d_ws != nullptr && ws_size >= wtBytes) {
    float* Wt = (float*)d_ws;
    transpose_w<<<N_NODES * N_NODES, 256, 0, stream>>>(W, Wt);
    (void)hipFuncSetAttribute((const void*)wired_dag_wmma<true>,
                              hipFuncAttributeMaxDynamicSharedMemorySize,
                              (int)ldsBytes);
    wired_dag_wmma<true><<<nBlocks, 128, ldsBytes, stream>>>(x, Wt, b, out);
  } else {
    (void)hipFuncSetAttribute((const void*)wired_dag_wmma<false>,
                              hipFuncAttributeMaxDynamicSharedMemorySize,
                              (int)ldsBytes);
    wired_dag_wmma<false><<<nBlocks, 128, ldsBytes, stream>>>(x, W, b, out);
  }
}